// EdgeWeightNorm_3796751090015
// MI455X (gfx1250) — compile-verified
//
#include <hip/hip_runtime.h>

// ---------------------------------------------------------------------------
// EdgeWeightNorm (GCN norm='both') for MI455X / gfx1250.
//
//   out_deg = segment_sum(w, src, N); in_deg = segment_sum(w, dst, N)
//   out[e]  = rsqrt(out_deg[src[e]]) * rsqrt(in_deg[dst[e]]) * w[e]
//
// Memory-bound scatter/gather: ~730 MB of HBM traffic => ~31 us floor at
// 23.3 TB/s. The 2*N*4B = 800 KB degree/norm table lives in the 192 MB L2
// (keep it RT); the 25.6M-edge streams are marked non-temporal so they do
// not rinse the hot table. Atomics are no-return global_atomic_add_f32.
// num_nodes is a device scalar (graph capture forbids sync readback), so all
// N-dependent arithmetic happens on-device.
// ---------------------------------------------------------------------------

typedef float f32x4 __attribute__((ext_vector_type(4)));
typedef int   i32x4 __attribute__((ext_vector_type(4)));

// ws layout: [0, N)      -> out-degree, then rsqrt(out-degree)
//            [N, 2N)     -> in-degree,  then rsqrt(in-degree)

__global__ void zero_deg_kernel(float* __restrict__ ws,
                                const int* __restrict__ num_nodes) {
    const int n2 = 2 * (*num_nodes);
    const int stride = gridDim.x * blockDim.x;
    for (int i = blockIdx.x * blockDim.x + threadIdx.x; i < n2; i += stride)
        ws[i] = 0.0f;
}

__global__ void degree_accum_kernel(const float* __restrict__ w,
                                    const int* __restrict__ src,
                                    const int* __restrict__ dst,
                                    float* __restrict__ ws,
                                    const int* __restrict__ num_nodes,
                                    int E) {
    float* __restrict__ deg_out = ws;
    float* __restrict__ deg_in  = ws + *num_nodes;

    const int base = (blockIdx.x * blockDim.x + threadIdx.x) * 4;
    if (base + 3 < E) {
        // 128-bit non-temporal streaming loads (edge arrays bypass-rinse L2;
        // the 800 KB degree table stays resident instead).
        f32x4 wv = __builtin_nontemporal_load((const f32x4*)(w   + base));
        i32x4 sv = __builtin_nontemporal_load((const i32x4*)(src + base));
        i32x4 dv = __builtin_nontemporal_load((const i32x4*)(dst + base));
#pragma unroll
        for (int k = 0; k < 4; ++k) {
            // Result discarded -> no-return global_atomic_add_f32 (STOREcnt,
            // resolved at the L2 atomic units; table is L2-resident).
            atomicAdd(deg_out + sv[k], wv[k]);
            atomicAdd(deg_in  + dv[k], wv[k]);
        }
    } else if (base < E) {
        for (int e = base; e < E; ++e) {
            const float we = w[e];
            atomicAdd(deg_out + src[e], we);
            atomicAdd(deg_in  + dst[e], we);
        }
    }
}

__global__ void rsqrt_inplace_kernel(float* __restrict__ ws,
                                     const int* __restrict__ num_nodes) {
    // eps == 0 in the reference: rsqrt(0) = +inf matches jax.lax.rsqrt.
    const int n2 = 2 * (*num_nodes);
    const int stride = gridDim.x * blockDim.x;
    for (int i = blockIdx.x * blockDim.x + threadIdx.x; i < n2; i += stride)
        ws[i] = rsqrtf(ws[i]);   // v_rsq_f32
}

__global__ void edge_norm_kernel(const float* __restrict__ w,
                                 const int* __restrict__ src,
                                 const int* __restrict__ dst,
                                 const float* __restrict__ ws,
                                 const int* __restrict__ num_nodes,
                                 float* __restrict__ out,
                                 int E) {
    const float* __restrict__ nsrc = ws;                // rsqrt(out_deg)
    const float* __restrict__ ndst = ws + *num_nodes;   // rsqrt(in_deg)

    const int base = (blockIdx.x * blockDim.x + threadIdx.x) * 4;
    if (base + 3 < E) {
        f32x4 wv = __builtin_nontemporal_load((const f32x4*)(w   + base));
        i32x4 sv = __builtin_nontemporal_load((const i32x4*)(src + base));
        i32x4 dv = __builtin_nontemporal_load((const i32x4*)(dst + base));
        f32x4 o;
#pragma unroll
        for (int k = 0; k < 4; ++k) {
            // Regular-temporal gathers: 800 KB table hits L2 (192 MB).
            o[k] = nsrc[sv[k]] * ndst[dv[k]] * wv[k];
        }
        // Streaming 128-bit NT store of the 102 MB result.
        __builtin_nontemporal_store(o, (f32x4*)(out + base));
    } else if (base < E) {
        for (int e = base; e < E; ++e)
            out[e] = nsrc[src[e]] * ndst[dst[e]] * w[e];
    }
}

extern "C" void kernel_launch(void* const* d_in, const int* in_sizes, int n_in,
                              void* d_out, int out_size, void* d_ws, size_t ws_size,
                              hipStream_t stream) {
    const float* w         = (const float*)d_in[0];
    const int*   src       = (const int*)d_in[1];
    const int*   dst       = (const int*)d_in[2];
    const int*   num_nodes = (const int*)d_in[3];   // device scalar
    float*       out       = (float*)d_out;
    float*       ws        = (float*)d_ws;          // needs 2*N*4 = 800 KB

    const int E = in_sizes[0];

    const int threads  = 256;                       // 8 wave32 waves / block
    const int nvec     = (E + 3) / 4;               // 4 edges per lane
    const int eBlocks  = (nvec + threads - 1) / threads;
    const int nBlocks  = 1024;                      // grid-stride over 2*N

    zero_deg_kernel<<<nBlocks, threads, 0, stream>>>(ws, num_nodes);
    degree_accum_kernel<<<eBlocks, threads, 0, stream>>>(w, src, dst, ws,
                                                         num_nodes, E);
    rsqrt_inplace_kernel<<<nBlocks, threads, 0, stream>>>(ws, num_nodes);
    edge_norm_kernel<<<eBlocks, threads, 0, stream>>>(w, src, dst, ws,
                                                      num_nodes, out, E);
}